// WA_1Conv_22943715295613
// MI455X (gfx1250) — compile-verified
//
#include <hip/hip_runtime.h>
#include <math.h>

#define B 32
#define INP 256
#define OUP 256
#define FGROUP 8
#define CG 32
#define HW 4096
#define HH 64
#define WW 64

#define KCHUNK 8            // K rows staged per TDM transfer
#define PBLK 64             // p columns per block
#define LROW 80             // LDS row stride in floats (64 data + 16 TDM pad)

typedef float v8f __attribute__((ext_vector_type(8)));
typedef float v2f __attribute__((ext_vector_type(2)));
typedef unsigned int u32;
typedef u32 u32x4 __attribute__((ext_vector_type(4)));
typedef u32 u32x8 __attribute__((ext_vector_type(8)));

// ---------------- 1. g = mean over H*W ----------------
__global__ void k_mean_hw(const float* __restrict__ x, float* __restrict__ g) {
    __shared__ float red[256];
    const int bc = blockIdx.x;              // b*INP + c
    const float* p = x + (size_t)bc * HW;
    float s = 0.f;
    for (int i = threadIdx.x; i < HW; i += 256) s += p[i];
    red[threadIdx.x] = s;
    __syncthreads();
    for (int off = 128; off > 0; off >>= 1) {
        if (threadIdx.x < off) red[threadIdx.x] += red[threadIdx.x + off];
        __syncthreads();
    }
    if (threadIdx.x == 0) g[bc] = red[0] * (1.f / (float)HW);
}

// ---------------- 2. GLGF -> s -> Wdyn ----------------
__global__ void k_glgf(const float* __restrict__ g,
                       const float* __restrict__ w1l,   // (8,32,32)
                       const float* __restrict__ w1g1,  // (8,32)
                       const float* __restrict__ w1g2,  // (256,8)
                       const float* __restrict__ w2l,   // (256,256)
                       const float* __restrict__ w2g,   // (256,)
                       float* __restrict__ Wdyn) {      // (B,256,256)
    __shared__ float sg[INP];
    __shared__ float t1s[FGROUP];
    __shared__ float ss[INP];
    __shared__ float red[INP];
    const int b = blockIdx.x;
    const int t = threadIdx.x;
    sg[t] = g[b * INP + t];
    __syncthreads();

    const int grp = t >> 5;
    const int o   = t & 31;
    float loc = 0.f;
    #pragma unroll 8
    for (int i = 0; i < CG; ++i)
        loc += sg[grp * CG + i] * w1l[grp * CG * CG + o * CG + i];

    if (t < FGROUP) {
        float v = 0.f;
        #pragma unroll 8
        for (int i = 0; i < CG; ++i) v += sg[t * CG + i] * w1g1[t * CG + i];
        t1s[t] = v;
    }
    __syncthreads();

    float glob = 0.f;
    #pragma unroll
    for (int gi = 0; gi < FGROUP; ++gi) glob += t1s[gi] * w1g2[t * FGROUP + gi];

    const float sv = 1.f / (1.f + expf(-(loc + glob)));
    ss[t] = sv;
    red[t] = sv * w2g[t];
    __syncthreads();
    for (int off = 128; off > 0; off >>= 1) {
        if (t < off) red[t] += red[t + off];
        __syncthreads();
    }
    const float sdot = red[0];

    float* Wb = Wdyn + (size_t)b * INP * OUP;
    for (int row = 0; row < OUP; ++row)
        Wb[row * INP + t] = ss[row] * w2l[row * INP + t] + sdot;
}

// ---------------- TDM: 2-D tile (tile_d1 rows x tile_d0 elems, fp32) -> LDS ----------------
// D# per CDNA5 ISA. Pad: 16 DWORDs after every 64 DWORDs
// => LDS row stride 80 floats, making the wave's two 16-lane halves bank-disjoint.
__device__ __forceinline__ void tdm_load_x_chunk(u32 lds_off, const float* gaddr) {
    const unsigned long long ga = (unsigned long long)(uintptr_t)gaddr;
    u32x4 g0;
    g0[0] = 1u;                                            // count=1 (valid), user mode
    g0[1] = lds_off;                                       // lds_addr (bytes)
    g0[2] = (u32)ga;                                       // global_addr[31:0]
    g0[3] = ((u32)(ga >> 32) & 0x01FFFFFFu) | 0x80000000u; // global_addr[56:32] | type=2
    u32x8 g1;
    g1[0] = (2u << 16)          // data_size = 4 bytes
          | (1u << 20)          // pad_enable
          | (5u << 22)          // pad_interval: 64 DWORDs
          | (15u << 25);        // pad_amount:  16 DWORDs
    g1[1] = ((u32)HW & 0xFFFFu) << 16;                 // tensor_dim0[15:0] (=4096)
    g1[2] = ((u32)HW >> 16) | (((u32)INP & 0xFFFFu) << 16); // dim0 hi | tensor_dim1 lo
    g1[3] = ((u32)INP >> 16) | ((u32)PBLK << 16);      // dim1 hi | tile_dim0 = 64
    g1[4] = (u32)KCHUNK;                               // tile_dim1 = 8, tile_dim2 = 0
    g1[5] = (u32)HW;                                   // tensor_dim0_stride[31:0]
    g1[6] = 0u;                                        // stride hi | dim1_stride lo
    g1[7] = 0u;
    asm volatile("tensor_load_to_lds %0, %1" :: "s"(g0), "s"(g1) : "memory");
}

// ---------------- 3. per-batch GEMM: Y = Wdyn * X via V_WMMA_F32_16X16X4_F32 ----------------
// One block covers ALL 256 output channels for a 64-wide p slab: each X element is
// moved HBM->LDS exactly once (by the TDM), then consumed by all 8 waves.
// __launch_bounds__(256, 1): single workgroup per EU target -> full VGPR budget,
// keeps all 8 f32 accumulator tiles + pipelined operands resident (no scratch spills).
__global__ void __launch_bounds__(256, 1)
k_gemm_wmma(const float* __restrict__ x,
            const float* __restrict__ Wdyn,
            float* __restrict__ y) {
    __shared__ float xbuf[2][KCHUNK * LROW];   // 2 x 2.5 KB double buffer (TDM-padded rows)

    const int b     = blockIdx.x;
    const int p_blk = blockIdx.y * PBLK;
    const int wv    = threadIdx.x >> 5;
    const int lane  = threadIdx.x & 31;
    const int o_base = wv * 32;                // 8 waves x 32 rows = all 256 channels

    const int m    = lane & 15;
    const int hi   = lane >> 4;
    const int koff = hi * 2;                   // lanes 16-31 carry K+2, K+3

    const float* __restrict__ Wb = Wdyn + (size_t)b * INP * OUP;
    const float* __restrict__ Xb = x    + (size_t)b * INP * HW + p_blk;
    float*       __restrict__ Yb = y    + (size_t)b * OUP * HW;

    // Per-wave A row base pointers (row-major Wdyn, K contiguous -> natural b64 loads)
    const float* __restrict__ Wr0 = Wb + (size_t)(o_base      + m) * INP + koff;
    const float* __restrict__ Wr1 = Wb + (size_t)(o_base + 16 + m) * INP + koff;

    const u32 lds0 = (u32)(uintptr_t)&xbuf[0][0];
    const u32 lds1 = (u32)(uintptr_t)&xbuf[1][0];

    v8f acc[2][4];
    #pragma unroll
    for (int gt = 0; gt < 2; ++gt)
        #pragma unroll
        for (int t = 0; t < 4; ++t)
            acc[gt][t] = (v8f)0.f;

    const int nch = INP / KCHUNK;              // 32 chunks
    if (wv == 0) tdm_load_x_chunk(lds0, Xb);   // prologue: chunk 0

    for (int i = 0; i < nch; ++i) {
        if (wv == 0) {
            if (i + 1 < nch) {
                tdm_load_x_chunk((i & 1) ? lds0 : lds1,
                                 Xb + (size_t)(i + 1) * KCHUNK * HW);
                __builtin_amdgcn_s_wait_tensorcnt(1);  // chunk i done (TDM in-order)
            } else {
                __builtin_amdgcn_s_wait_tensorcnt(0);
            }
        }
        __syncthreads();                        // publish buffer i to all waves

        const float* xb = &xbuf[i & 1][0];
        #pragma unroll
        for (int kk = 0; kk < KCHUNK; kk += 4) {
            const int k = i * KCHUNK + kk;
            v2f a0 = *(const v2f*)(Wr0 + k);
            v2f a1 = *(const v2f*)(Wr1 + k);
            const float* xr0 = xb + (kk + koff) * LROW + m;
            const float* xr1 = xr0 + LROW;
            #pragma unroll
            for (int t = 0; t < 4; ++t) {
                v2f bb;
                bb.x = xr0[t * 16];
                bb.y = xr1[t * 16];
                acc[0][t] = __builtin_amdgcn_wmma_f32_16x16x4_f32(
                    false, a0, false, bb, (short)0, acc[0][t], false, false);
                acc[1][t] = __builtin_amdgcn_wmma_f32_16x16x4_f32(
                    false, a1, false, bb, (short)0, acc[1][t], false, false);
            }
        }
        __syncthreads();                        // all waves done before buffer reuse
    }

    // C/D layout: VGPR r -> rows M=r (lanes 0-15) and M=r+8 (lanes 16-31)
    #pragma unroll
    for (int gt = 0; gt < 2; ++gt) {
        #pragma unroll
        for (int t = 0; t < 4; ++t) {
            #pragma unroll
            for (int r = 0; r < 8; ++r) {
                const int row = o_base + gt * 16 + r + hi * 8;
                const int col = p_blk + t * 16 + m;
                Yb[(size_t)row * HW + col] = acc[gt][t][r];
            }
        }
    }
}

// ---------------- 4. channel mean ----------------
__global__ void k_cmean(const float* __restrict__ y, float* __restrict__ cmap) {
    const int idx = blockIdx.x * 256 + threadIdx.x;   // b*HW + p
    const int b = idx >> 12;
    const int p = idx & (HW - 1);
    const float* yb = y + (size_t)b * OUP * HW + p;
    float s = 0.f;
    for (int o = 0; o < OUP; ++o) s += yb[(size_t)o * HW];
    cmap[idx] = s * (1.f / (float)OUP);
}

// ---------------- 5/6. 3x3 conv (cross-correlation, zero pad), optional sigmoid ----------------
template <bool SIG>
__global__ void k_conv3(const float* __restrict__ in, const float* __restrict__ w9,
                        float* __restrict__ out) {
    const int idx = blockIdx.x * 256 + threadIdx.x;   // b*HW + p
    const int b = idx >> 12;
    const int p = idx & (HW - 1);
    const int h = p >> 6;
    const int w = p & 63;
    const float* ib = in + (size_t)b * HW;
    float acc = 0.f;
    #pragma unroll
    for (int kh = 0; kh < 3; ++kh) {
        const int hh = h + kh - 1;
        if (hh < 0 || hh >= HH) continue;
        #pragma unroll
        for (int kw = 0; kw < 3; ++kw) {
            const int ww = w + kw - 1;
            if (ww < 0 || ww >= WW) continue;
            acc += ib[hh * WW + ww] * w9[kh * 3 + kw];
        }
    }
    if (SIG) acc = 1.f / (1.f + expf(-acc));
    out[idx] = acc;
}

// ---------------- 7. gating ----------------
__global__ void k_scale(float* __restrict__ y, const float* __restrict__ cmap) {
    const size_t idx = (size_t)blockIdx.x * 256 + threadIdx.x;  // over B*OUP*HW
    const int b = (int)(idx >> 20);                              // OUP*HW = 1<<20
    const int p = (int)(idx & (HW - 1));
    y[idx] *= cmap[b * HW + p];
}

extern "C" void kernel_launch(void* const* d_in, const int* in_sizes, int n_in,
                              void* d_out, int out_size, void* d_ws, size_t ws_size,
                              hipStream_t stream) {
    const float* x       = (const float*)d_in[0];
    const float* w1l     = (const float*)d_in[1];
    const float* w1g1    = (const float*)d_in[2];
    const float* w1g2    = (const float*)d_in[3];
    const float* w2l     = (const float*)d_in[4];
    const float* w2g     = (const float*)d_in[5];
    const float* conv1_w = (const float*)d_in[6];
    const float* conv2_w = (const float*)d_in[7];
    float* out = (float*)d_out;

    float* ws    = (float*)d_ws;
    float* g     = ws;                             // B*INP
    float* Wdyn  = g + B * INP;                    // B*INP*OUP
    float* cmapA = Wdyn + (size_t)B * INP * OUP;   // B*HW
    float* cmapB = cmapA + B * HW;
    float* cmapC = cmapB + B * HW;

    k_mean_hw<<<B * INP, 256, 0, stream>>>(x, g);
    k_glgf<<<B, 256, 0, stream>>>(g, w1l, w1g1, w1g2, w2l, w2g, Wdyn);
    {
        dim3 grid(B, HW / PBLK);                   // 32 x 64 blocks, 8 waves each
        k_gemm_wmma<<<grid, 256, 0, stream>>>(x, Wdyn, out);
    }
    k_cmean<<<(B * HW) / 256, 256, 0, stream>>>(out, cmapA);
    k_conv3<true ><<<(B * HW) / 256, 256, 0, stream>>>(cmapA, conv1_w, cmapB);
    k_conv3<false><<<(B * HW) / 256, 256, 0, stream>>>(cmapB, conv2_w, cmapC);
    k_scale<<<(B * OUP * HW) / 256, 256, 0, stream>>>(out, cmapC);
}